// GraphNetwork_46634754900621
// MI455X (gfx1250) — compile-verified
//
#include <hip/hip_runtime.h>
#include <hip/hip_bf16.h>
#include <math.h>

typedef __attribute__((ext_vector_type(2))) float v2f;
typedef __attribute__((ext_vector_type(8))) float v8f;

#define D_MODEL    128
#define NHEAD      8
#define HEAD_DIM   16
#define NUM_LAYERS 7
#define NUM_LABELS 2
#define N_NODES    10000
#define N_EDGES    160000
#define E_TOT      (N_EDGES + N_NODES)   // with self loops
#define NEG_SLOPE  0.2f

// ---------- order-preserving float <-> uint encoding for atomic segment-max ----------
__device__ __forceinline__ unsigned fenc(float f) {
    unsigned u = __float_as_uint(f);
    return (u & 0x80000000u) ? ~u : (u | 0x80000000u);
}
__device__ __forceinline__ float fdec(unsigned u) {
    unsigned b = (u & 0x80000000u) ? (u & 0x7FFFFFFFu) : ~u;
    return __uint_as_float(b);
}
#define ENC_NEG_INF 0x007FFFFFu   // fenc(-inf)

// ---------- fused dual projection: Yl = X@Wl + bl ; Yr = X@Wr + br (fp32 WMMA) ----------
// grid = (N/16, 2), block = 256 (8 waves). Wave w computes 16x16 tile at cols [16w,16w+16).
__global__ __launch_bounds__(256)
void gemm128_wmma(const float* __restrict__ X,
                  const float* __restrict__ Wl, const float* __restrict__ bl,
                  const float* __restrict__ Wr, const float* __restrict__ br,
                  float* __restrict__ Yl, float* __restrict__ Yr) {
    const float* W  = blockIdx.y ? Wr : Wl;
    const float* bv = blockIdx.y ? br : bl;
    float*       Y  = blockIdx.y ? Yr : Yl;

    const int wave = threadIdx.x >> 5;
    const int lane = threadIdx.x & 31;
    const int row0 = blockIdx.x * 16;
    const int col0 = wave * 16;
    const int m    = lane & 15;   // row (A) / col (B,D) within tile
    const int kh   = lane >> 4;   // half-wave selects K pair {0,1} vs {2,3}

    const float* xrow = X + (size_t)(row0 + m) * D_MODEL;
    v8f acc = {};
    #pragma unroll 4
    for (int k = 0; k < D_MODEL; k += 4) {
        // A 16x4 f32: VGPR0 = K even-of-pair, VGPR1 = K odd-of-pair (ISA 7.12.2)
        v2f a, b;
        a.x = xrow[k + kh * 2 + 0];
        a.y = xrow[k + kh * 2 + 1];
        // B 4x16 f32: lane = N column, same K-pair striping
        b.x = W[(size_t)(k + kh * 2 + 0) * D_MODEL + col0 + m];
        b.y = W[(size_t)(k + kh * 2 + 1) * D_MODEL + col0 + m];
        acc = __builtin_amdgcn_wmma_f32_16x16x4_f32(
                  false, a, false, b, (short)0, acc, false, false);
    }
    // C/D 16x16 f32 layout: VGPR v -> M = v + (half-wave ? 8 : 0), lane&15 -> N
    const int ncol  = col0 + m;
    const float bia = bv[ncol];
    const int mbase = row0 + (kh ? 8 : 0);
    #pragma unroll
    for (int v = 0; v < 8; v++)
        Y[(size_t)(mbase + v) * D_MODEL + ncol] = acc[v] + bia;
}

// ---------- per-layer reset of segment-max / denom / output accumulator ----------
__global__ __launch_bounds__(256)
void layer_init(unsigned* __restrict__ emax, float* __restrict__ denom,
                float* __restrict__ outacc) {
    int t = blockIdx.x * 256 + threadIdx.x;
    if (t < N_NODES * NHEAD) { emax[t] = ENC_NEG_INF; denom[t] = 0.f; }
    if (t < N_NODES * D_MODEL) outacc[t] = 0.f;
}

__device__ __forceinline__ void edge_sd(const int* __restrict__ eidx, int e, int& s, int& d) {
    if (e < N_EDGES) { s = eidx[e]; d = eidx[N_EDGES + e]; }
    else             { s = d = e - N_EDGES; }          // self loop
}

// ---------- pass 1: e = att . leaky_relu(xl[src] + xr[dst]) ; segment max ----------
__global__ __launch_bounds__(256)
void edge_score(const float* __restrict__ xl, const float* __restrict__ xr,
                const int* __restrict__ eidx, const float* __restrict__ att,
                float* __restrict__ esc, unsigned* __restrict__ emax) {
    int t = blockIdx.x * 256 + threadIdx.x;
    if (t >= E_TOT * NHEAD) return;
    int e = t >> 3, h = t & 7, s, d;
    edge_sd(eidx, e, s, d);
    const float4* pl = (const float4*)(xl + (size_t)s * D_MODEL + h * HEAD_DIM);
    const float4* pr = (const float4*)(xr + (size_t)d * D_MODEL + h * HEAD_DIM);
    const float4* pa = (const float4*)(att + h * HEAD_DIM);
    float acc = 0.f;
    #pragma unroll
    for (int i = 0; i < 4; i++) {
        float4 a = pl[i], b = pr[i], w = pa[i];
        float z0 = a.x + b.x, z1 = a.y + b.y, z2 = a.z + b.z, z3 = a.w + b.w;
        z0 = z0 > 0.f ? z0 : z0 * NEG_SLOPE;
        z1 = z1 > 0.f ? z1 : z1 * NEG_SLOPE;
        z2 = z2 > 0.f ? z2 : z2 * NEG_SLOPE;
        z3 = z3 > 0.f ? z3 : z3 * NEG_SLOPE;
        acc += z0 * w.x + z1 * w.y + z2 * w.z + z3 * w.w;
    }
    esc[t] = acc;
    atomicMax(emax + (size_t)d * NHEAD + h, fenc(acc));
}

// ---------- pass 2 (fused): ex = exp(e - max); denom += ex; out[dst] += ex * xl[src] ----------
// Normalization by denom is deferred to the node-level finalize (alpha = ex/denom is
// separable since denom is constant per (dst,head)).
__global__ __launch_bounds__(256)
void edge_accum(const int* __restrict__ eidx, const float* __restrict__ xl,
                const unsigned* __restrict__ emax, const float* __restrict__ esc,
                float* __restrict__ denom, float* __restrict__ outacc) {
    int t = blockIdx.x * 256 + threadIdx.x;
    if (t >= E_TOT * NHEAD) return;
    int e = t >> 3, h = t & 7, s, d;
    edge_sd(eidx, e, s, d);
    float m  = fdec(emax[(size_t)d * NHEAD + h]);
    float ex = __expf(esc[t] - m);
    atomicAdd(denom + (size_t)d * NHEAD + h, ex);
    const float4* pl = (const float4*)(xl + (size_t)s * D_MODEL + h * HEAD_DIM);
    float* po = outacc + (size_t)d * D_MODEL + h * HEAD_DIM;
    #pragma unroll
    for (int i = 0; i < 4; i++) {
        float4 a = pl[i];
        atomicAdd(po + i * 4 + 0, ex * a.x);
        atomicAdd(po + i * 4 + 1, ex * a.y);
        atomicAdd(po + i * 4 + 2, ex * a.z);
        atomicAdd(po + i * 4 + 3, ex * a.w);
    }
}

// ---------- x_next = outacc / (denom + 1e-16) + bias ----------
__global__ __launch_bounds__(256)
void finalize_norm(const float* __restrict__ outacc, const float* __restrict__ denom,
                   const float* __restrict__ bias, float* __restrict__ xout) {
    int t = blockIdx.x * 256 + threadIdx.x;
    if (t >= N_NODES * D_MODEL) return;
    int node = t >> 7;                 // /128
    int dim  = t & (D_MODEL - 1);
    int h    = dim >> 4;               // /16
    float dn = denom[(size_t)node * NHEAD + h] + 1e-16f;
    xout[t] = outacc[t] / dn + bias[dim];
}

// ---------- head: keep first NUM_LABELS rows of each chunk, 128->1 linear ----------
__global__ __launch_bounds__(256)
void head_kernel(const float* __restrict__ x, const int* __restrict__ nchunks,
                 const float* __restrict__ wh, const float* __restrict__ bh,
                 float* __restrict__ out, int out_size) {
    int t = blockIdx.x * 256 + threadIdx.x;
    if (t >= out_size) return;
    int chunk = NUM_LABELS + nchunks[0];
    int c = t / NUM_LABELS, l = t % NUM_LABELS;
    const float4* p = (const float4*)(x + (size_t)(c * chunk + l) * D_MODEL);
    const float4* w = (const float4*)wh;
    float acc = 0.f;
    #pragma unroll 8
    for (int i = 0; i < D_MODEL / 4; i++) {
        float4 a = p[i], b = w[i];
        acc += a.x * b.x + a.y * b.y + a.z * b.z + a.w * b.w;
    }
    out[t] = acc + bh[0];
}

extern "C" void kernel_launch(void* const* d_in, const int* in_sizes, int n_in,
                              void* d_out, int out_size, void* d_ws, size_t ws_size,
                              hipStream_t stream) {
    const float* x_in   = (const float*)d_in[0];
    const int*   eidx   = (const int*)d_in[1];
    const int*   nchnk  = (const int*)d_in[2];
    const float* Wl     = (const float*)d_in[3];
    const float* bl     = (const float*)d_in[4];
    const float* Wr     = (const float*)d_in[5];
    const float* br     = (const float*)d_in[6];
    const float* att    = (const float*)d_in[7];
    const float* bias   = (const float*)d_in[8];
    const float* w_head = (const float*)d_in[9];
    const float* b_head = (const float*)d_in[10];

    // workspace carve-up (all fp32 unless noted); ~32 MB total, L2-resident
    float* ws     = (float*)d_ws;
    float* bufA   = ws;
    float* bufB   = bufA   + (size_t)N_NODES * D_MODEL;
    float* xl     = bufB   + (size_t)N_NODES * D_MODEL;
    float* xr     = xl     + (size_t)N_NODES * D_MODEL;
    float* outacc = xr     + (size_t)N_NODES * D_MODEL;
    float* esc    = outacc + (size_t)N_NODES * D_MODEL;
    float* denom  = esc    + (size_t)E_TOT * NHEAD;
    unsigned* emax = (unsigned*)(denom + (size_t)N_NODES * NHEAD);

    const int nEdgeThreads = E_TOT * NHEAD;
    const int gEdge = (nEdgeThreads + 255) / 256;
    const int gNode = (N_NODES * D_MODEL + 255) / 256;

    const float* xcur = x_in;
    for (int i = 0; i < NUM_LAYERS; i++) {
        float* xnext = (i & 1) ? bufB : bufA;
        gemm128_wmma<<<dim3(N_NODES / 16, 2), 256, 0, stream>>>(
            xcur,
            Wl + (size_t)i * D_MODEL * D_MODEL, bl + (size_t)i * D_MODEL,
            Wr + (size_t)i * D_MODEL * D_MODEL, br + (size_t)i * D_MODEL,
            xl, xr);
        layer_init<<<gNode, 256, 0, stream>>>(emax, denom, outacc);
        edge_score<<<gEdge, 256, 0, stream>>>(xl, xr, eidx,
            att + (size_t)i * NHEAD * HEAD_DIM, esc, emax);
        edge_accum<<<gEdge, 256, 0, stream>>>(eidx, xl, emax, esc, denom, outacc);
        finalize_norm<<<gNode, 256, 0, stream>>>(outacc, denom,
            bias + (size_t)i * D_MODEL, xnext);
        xcur = xnext;
    }
    head_kernel<<<(out_size + 255) / 256, 256, 0, stream>>>(
        xcur, nchnk, w_head, b_head, (float*)d_out, out_size);
}